// Attn_3092376453484
// MI455X (gfx1250) — compile-verified
//
#include <hip/hip_runtime.h>
#include <hip/hip_bf16.h>
#include <math.h>

typedef float v2f __attribute__((ext_vector_type(2)));
typedef float v4f __attribute__((ext_vector_type(4)));
typedef float v8f __attribute__((ext_vector_type(8)));

static constexpr int Bb = 32;    // batch
static constexpr int Ll = 4096;  // time steps
static constexpr int Hh = 1024;  // hidden

// ---------------------------------------------------------------------------
// Kernel 1: v = hid @ W   (32x1024 = 32x1024 * 1024x1024), f32 WMMA 16x16x4.
// One wave per 16x16 output tile -> grid = (32/16)*(1024/16) = 128 waves.
// A-layout (16x4 f32): lanes 0-15 hold {K0,K1}, lanes 16-31 hold {K2,K3}.
// B-layout (4x16 f32): mirrored. C/D: VGPR r = row (r + 8*half), col = lane%16.
// ---------------------------------------------------------------------------
__global__ __launch_bounds__(32) void k_proj_wmma(
    const float* __restrict__ hid, const float* __restrict__ W,
    float* __restrict__ v_out) {
  const int tile = blockIdx.x;
  const int m0 = (tile & 1) << 4;        // batch-tile (0 or 16)
  const int n0 = (tile >> 1) << 4;       // output-feature tile
  const int lane = threadIdx.x & 31;
  const int half = lane >> 4;            // 0: lanes 0-15, 1: lanes 16-31
  const int q = lane & 15;

  const float* Arow = hid + (m0 + q) * Hh;   // hid[b = m0+q, :]
  v8f acc = {};
  for (int k0 = 0; k0 < Hh; k0 += 4) {
    const int ka = k0 + (half << 1);
    v2f a, b;
    a[0] = Arow[ka];
    a[1] = Arow[ka + 1];
    b[0] = W[(size_t)ka * Hh + n0 + q];        // W[g=ka,   h=n0+q]
    b[1] = W[(size_t)(ka + 1) * Hh + n0 + q];  // W[g=ka+1, h=n0+q]
    acc = __builtin_amdgcn_wmma_f32_16x16x4_f32(
        /*neg_a=*/false, a, /*neg_b=*/false, b,
        /*c_mod=*/(short)0, acc, /*reuse_a=*/false, /*reuse_b=*/false);
  }
  const int mb = m0 + (half << 3);
#pragma unroll
  for (int r = 0; r < 8; ++r)
    v_out[(size_t)(mb + r) * Hh + n0 + q] = acc[r];
}

// ---------------------------------------------------------------------------
// Kernel 1b: c[b] = hid[b] . bias   (tiny)
// ---------------------------------------------------------------------------
__global__ __launch_bounds__(256) void k_biasdot(
    const float* __restrict__ hid, const float* __restrict__ bias,
    float* __restrict__ c_out) {
  __shared__ float red[256];
  const int b = blockIdx.x;
  const int tid = threadIdx.x;
  float s = 0.f;
  for (int h = tid; h < Hh; h += 256) s += hid[b * Hh + h] * bias[h];
  red[tid] = s;
  __syncthreads();
  for (int off = 128; off > 0; off >>= 1) {
    if (tid < off) red[tid] += red[tid + off];
    __syncthreads();
  }
  if (tid == 0) c_out[b] = red[0];
}

// ---------------------------------------------------------------------------
// Kernel 2 (the 512 MiB streamer): energy[b,l] = v[b].enc[l,b] + c[b].
// One wave per (b, 8 consecutive l). v[b] (4KB) lives in 32 VGPRs and is
// reused across the 8 time steps; enc is streamed with nontemporal float4
// loads (it will never fit/stay in the 192MB L2, v and c will).
// ---------------------------------------------------------------------------
__global__ __launch_bounds__(256) void k_energies(
    const float* __restrict__ enc, const float* __restrict__ v,
    const float* __restrict__ c, float* __restrict__ energ) {
  const int wave = (blockIdx.x * 256 + threadIdx.x) >> 5;
  const int lane = threadIdx.x & 31;
  const int b = wave & (Bb - 1);
  const int l0 = (wave >> 5) << 3;  // 8 l's per wave

  const v4f* vp = (const v4f*)(v + (size_t)b * Hh);
  v4f vr[8];
#pragma unroll
  for (int i = 0; i < 8; ++i) vr[i] = vp[i * 32 + lane];

  const float cb = c[b];
#pragma unroll
  for (int j = 0; j < 8; ++j) {
    const int l = l0 + j;
    const v4f* ep = (const v4f*)(enc + (size_t)(l * Bb + b) * Hh);
    float s = 0.f;
#pragma unroll
    for (int i = 0; i < 8; ++i) {
      v4f e = __builtin_nontemporal_load(&ep[i * 32 + lane]);
      s += e[0] * vr[i][0] + e[1] * vr[i][1] + e[2] * vr[i][2] +
           e[3] * vr[i][3];
    }
#pragma unroll
    for (int off = 16; off > 0; off >>= 1) s += __shfl_xor(s, off, 32);
    if (lane == 0) energ[(size_t)b * Ll + l] = s + cb;
  }
}

// ---------------------------------------------------------------------------
// Kernel 3: softmax over L per batch row, in place on d_out. Row in LDS.
// ---------------------------------------------------------------------------
__global__ __launch_bounds__(256) void k_softmax(float* __restrict__ out) {
  __shared__ float sdata[Ll];  // 16 KB of the 320 KB WGP LDS
  __shared__ float red[256];
  const int b = blockIdx.x;
  const int tid = threadIdx.x;
  float* row = out + (size_t)b * Ll;

  float lmax = -INFINITY;
  for (int i = tid; i < Ll; i += 256) {
    float e = row[i];
    sdata[i] = e;
    lmax = fmaxf(lmax, e);
  }
  red[tid] = lmax;
  __syncthreads();
  for (int off = 128; off > 0; off >>= 1) {
    if (tid < off) red[tid] = fmaxf(red[tid], red[tid + off]);
    __syncthreads();
  }
  const float m = red[0];
  __syncthreads();

  float lsum = 0.f;
  for (int i = tid; i < Ll; i += 256) {
    float e = __expf(sdata[i] - m);
    sdata[i] = e;
    lsum += e;
  }
  red[tid] = lsum;
  __syncthreads();
  for (int off = 128; off > 0; off >>= 1) {
    if (tid < off) red[tid] += red[tid + off];
    __syncthreads();
  }
  const float inv = 1.0f / red[0];
  for (int i = tid; i < Ll; i += 256) row[i] = sdata[i] * inv;
}

// ---------------------------------------------------------------------------
extern "C" void kernel_launch(void* const* d_in, const int* in_sizes, int n_in,
                              void* d_out, int out_size, void* d_ws,
                              size_t ws_size, hipStream_t stream) {
  const float* hid  = (const float*)d_in[0];  // (1,32,1024)
  const float* enc  = (const float*)d_in[1];  // (4096,32,1024)
  const float* W    = (const float*)d_in[2];  // (1024,1024)
  const float* bias = (const float*)d_in[3];  // (1024,)
  float* out = (float*)d_out;                 // (32,1,4096)

  float* v = (float*)d_ws;                    // 32*1024 floats
  float* c = v + Bb * Hh;                     // 32 floats

  // 1) v = hid @ W  (WMMA f32 16x16x4; 128 single-wave tiles)
  k_proj_wmma<<<(Bb / 16) * (Hh / 16), 32, 0, stream>>>(hid, W, v);
  // 1b) c[b] = hid[b].bias
  k_biasdot<<<Bb, 256, 0, stream>>>(hid, bias, c);
  // 2) energies -> d_out  (bandwidth-bound: streams 512 MiB of enc)
  k_energies<<<(Bb * Ll) / (8 * 8), 256, 0, stream>>>(enc, v, c, out);
  // 3) softmax over time, in place
  k_softmax<<<Bb, 256, 0, stream>>>(out);
}